// MyCrossEntropy_90013924589968
// MI455X (gfx1250) — compile-verified
//
#include <hip/hip_runtime.h>
#include <hip/hip_bf16.h>
#include <math.h>

// Cross-entropy (mean of -log_softmax(pred)[i, y[i]]) for pred[B, V] f32.
// Pure streaming workload (~823 MB read once -> ~35us floor @ 23.3 TB/s).
// Uses the CDNA5 async global->LDS pipeline (global_load_async_to_lds_b128,
// ASYNCcnt) with a 4-deep LDS ring buffer and non-temporal cache hints,
// online-softmax accumulation, wave32 shuffle reductions.

#define CE_BLOCK 512   // 16 waves of 32
#define CE_NBUF  4     // LDS ring depth
#define CE_PF    3     // tiles in flight beyond the one being consumed

__device__ __forceinline__ void online_combine(float& m, float& s, float om, float os) {
    float nm = fmaxf(m, om);
    s = s * __expf(m - nm) + os * __expf(om - nm);
    m = nm;
}

__device__ __forceinline__ void online_accum(float& m, float& s, float x) {
    if (x > m) { s *= __expf(m - x); m = x; }
    s += __expf(x - m);
}

__global__ __launch_bounds__(CE_BLOCK)
void ce_row_kernel(const float* __restrict__ pred,
                   const int* __restrict__ y,
                   float* __restrict__ row_loss,
                   int vocab) {
    const int b   = blockIdx.x;
    const int tid = threadIdx.x;
    const float* __restrict__ row = pred + (size_t)b * (size_t)vocab;

    __shared__ float4 tile[CE_NBUF][CE_BLOCK];   // 32 KB ring buffer
    __shared__ float  red_m[CE_BLOCK / 32];
    __shared__ float  red_s[CE_BLOCK / 32];

    // Rows are only 4-byte aligned in general; peel to 16-byte alignment.
    const unsigned misal = (unsigned)((uintptr_t)row & 15u);
    const int lead = (int)(((16u - misal) & 15u) >> 2);        // 0..3 floats
    const int n4 = (vocab - lead) >> 2;                        // aligned float4 count
    const int tail_start = lead + (n4 << 2);
    const float4* __restrict__ main4 = (const float4*)(row + lead);
    const int num_tiles = (n4 + CE_BLOCK - 1) / CE_BLOCK;

    float m = -INFINITY;
    float s = 0.0f;

    // Every lane always issues (clamped address) so each wave's ASYNCcnt
    // bookkeeping is uniform and the scalar wait immediates are exact.
    auto issue_tile = [&](int t) {
        int idx = t * CE_BLOCK + tid; if (idx >= n4) idx = n4 - 1;
        unsigned long long ga = (unsigned long long)(uintptr_t)(main4 + idx);
        unsigned lo = (unsigned)(uintptr_t)(&tile[t & (CE_NBUF - 1)][tid]);
        asm volatile("global_load_async_to_lds_b128 %0, %1, off th:TH_LOAD_NT"
                     :: "v"(lo), "v"(ga) : "memory");
    };

    auto consume_tile = [&](int t) {
        const int idx = t * CE_BLOCK + tid;
        if (idx < n4) {
            float4 v = tile[t & (CE_NBUF - 1)][tid];           // ds_load_b128
            float c = fmaxf(fmaxf(v.x, v.y), fmaxf(v.z, v.w));
            if (c > m) { s *= __expf(m - c); m = c; }          // rescale (rare)
            s += __expf(v.x - m) + __expf(v.y - m) +
                 __expf(v.z - m) + __expf(v.w - m);
        }
    };

    // ---- prologue: fill the pipeline with up to CE_PF+0.. tiles ----
    const int npre = (num_tiles < CE_PF) ? num_tiles : CE_PF;
    for (int t = 0; t < npre; ++t) issue_tile(t);

    // ---- steady state: CE_PF+1 loads outstanding after each issue ----
    int t = 0;
    for (; t + CE_PF < num_tiles; ++t) {
        issue_tile(t + CE_PF);
        asm volatile("s_wait_asynccnt 3" ::: "memory");        // tile t landed
        consume_tile(t);
    }

    // ---- epilogue: everything remaining is already issued; drain once ----
    asm volatile("s_wait_asynccnt 0" ::: "memory");
    for (; t < num_tiles; ++t) consume_tile(t);

    // ---- lead + tail scalars (<= 6 elements total), thread 0 only ----
    if (tid == 0) {
        for (int i = 0; i < lead; ++i)            online_accum(m, s, row[i]);
        for (int i = tail_start; i < vocab; ++i)  online_accum(m, s, row[i]);
    }

    // ---- wave32 reduction (every lane holds finite m: n4 >> CE_BLOCK) ----
    #pragma unroll
    for (int off = 16; off >= 1; off >>= 1) {
        float om = __shfl_xor(m, off, 32);
        float os = __shfl_xor(s, off, 32);
        online_combine(m, s, om, os);
    }

    const int wave = tid >> 5;
    const int lane = tid & 31;
    if (lane == 0) { red_m[wave] = m; red_s[wave] = s; }
    __syncthreads();

    if (wave == 0) {
        const int nwaves = CE_BLOCK >> 5;                      // 16
        m = (lane < nwaves) ? red_m[lane] : -INFINITY;
        s = (lane < nwaves) ? red_s[lane] : 0.0f;
        // off=16 runs first, so every -inf lane pairs with a finite one
        // before smaller strides execute.
        #pragma unroll
        for (int off = 16; off >= 1; off >>= 1) {
            float om = __shfl_xor(m, off, 32);
            float os = __shfl_xor(s, off, 32);
            online_combine(m, s, om, os);
        }
        if (lane == 0) {
            float tlogit = row[y[b]];
            row_loss[b] = (logf(s) + m) - tlogit;              // -log softmax
        }
    }
}

__global__ __launch_bounds__(256)
void ce_reduce_kernel(const float* __restrict__ row_loss,
                      float* __restrict__ out,
                      int n, float inv_n) {
    __shared__ float sdata[256 / 32];
    float s = 0.0f;
    for (int i = threadIdx.x; i < n; i += 256) s += row_loss[i];

    #pragma unroll
    for (int off = 16; off >= 1; off >>= 1) s += __shfl_xor(s, off, 32);

    const int wave = threadIdx.x >> 5;
    const int lane = threadIdx.x & 31;
    if (lane == 0) sdata[wave] = s;
    __syncthreads();

    if (wave == 0) {
        s = (lane < (256 / 32)) ? sdata[lane] : 0.0f;
        #pragma unroll
        for (int off = 4; off >= 1; off >>= 1) s += __shfl_xor(s, off, 32);
        if (lane == 0) out[0] = s * inv_n;
    }
}

extern "C" void kernel_launch(void* const* d_in, const int* in_sizes, int n_in,
                              void* d_out, int out_size, void* d_ws, size_t ws_size,
                              hipStream_t stream) {
    const float* pred = (const float*)d_in[0];
    const int*   y    = (const int*)d_in[1];
    float*       out  = (float*)d_out;
    float*       ws   = (float*)d_ws;   // batch floats of per-row loss

    const int batch = in_sizes[1];
    const int vocab = (int)(in_sizes[0] / (long long)batch);

    ce_row_kernel<<<batch, CE_BLOCK, 0, stream>>>(pred, y, ws, vocab);
    ce_reduce_kernel<<<1, 256, 0, stream>>>(ws, out, batch, 1.0f / (float)batch);
}